// GNN_45105746543056
// MI455X (gfx1250) — compile-verified
//
#include <hip/hip_runtime.h>

#define DFEAT 64

typedef __attribute__((ext_vector_type(2))) float v2f;
typedef __attribute__((ext_vector_type(8))) float v8f;

// ---------------------------------------------------------------------------
// Zero-fill (agg buffer must be zeroed every call; harness poisons ws once)
// ---------------------------------------------------------------------------
__global__ void gnn_zero_kernel(float* __restrict__ p, int n) {
    int i = blockIdx.x * blockDim.x + threadIdx.x;
    if (i < n) p[i] = 0.0f;
}

// ---------------------------------------------------------------------------
// Edge scatter-add: agg[dst[e]] += xin[src[e]]   (16 threads per edge,
// each thread moves one float4 and issues 4 global_atomic_add_f32; the
// 25.6MB agg buffer is L2-resident so the RMW happens in L2)
// ---------------------------------------------------------------------------
__global__ void gnn_scatter_add_kernel(const float* __restrict__ xin,
                                       const int* __restrict__ src,
                                       const int* __restrict__ dst,
                                       float* __restrict__ agg,
                                       int n_edges) {
    int t = blockIdx.x * blockDim.x + threadIdx.x;
    int e = t >> 4;                  // 16 threads per edge
    if (e >= n_edges) return;
    int c = (t & 15) << 2;           // float4 chunk within the 64-wide row
    int s = src[e];
    int d = dst[e];
    const float4 v = *reinterpret_cast<const float4*>(xin + (size_t)s * DFEAT + c);
    float* o = agg + (size_t)d * DFEAT + c;
    atomicAdd(o + 0, v.x);
    atomicAdd(o + 1, v.y);
    atomicAdd(o + 2, v.z);
    atomicAdd(o + 3, v.w);
}

// ---------------------------------------------------------------------------
// Fused GraphConv GEMM:
//   out[m][n] = sum_k [agg|x][m][k] * [W_rel;W_root][n][k] + bias[n]  (ReLU opt)
// One 128-thread block (4 waves) computes a 16x64 output tile.
// Wave w owns output columns [16w, 16w+16); K=128 via 32x V_WMMA_F32_16X16X4_F32.
//
// fp32 WMMA lane layout (wave32):
//   A 16x4 : lanes 0-15 row M=lane, v0/v1 = K+0/K+1 ; lanes 16-31 = K+2/K+3
//   B 4x16 : lanes 0-15 col N=lane, v0/v1 = K+0/K+1 ; lanes 16-31 = K+2/K+3
//   C/D    : VGPR r -> M = r + 8*(lane>=16), N = lane&15
// LDS row stride 132 => the two half-wave b64 reads hit disjoint bank sets.
// ---------------------------------------------------------------------------
template <bool RELU>
__global__ void __launch_bounds__(128, 1)
gnn_gconv_gemm_kernel(const float* __restrict__ agg,    // [N,64]
                      const float* __restrict__ xin,    // [N,64]
                      const float* __restrict__ Wrel,   // [64,64] row-major
                      const float* __restrict__ Wroot,  // [64,64] row-major
                      const float* __restrict__ bias,   // [64]
                      float* __restrict__ out,          // [N,64]
                      int n_nodes) {
    __shared__ float As[16][132];   // [node-in-tile][k]  k: 0-63 agg, 64-127 x
    __shared__ float Bs[64][132];   // [n][k]             k: 0-63 Wrel, 64-127 Wroot

    const int tid = threadIdx.x;
    const int rb  = blockIdx.x * 16;

    // --- stage A tile (clamped rows keep EXEC fully enabled for WMMA) ---
#pragma unroll
    for (int r = 0; r < 16; ++r) {
        int row = rb + r;
        if (row >= n_nodes) row = n_nodes - 1;
        float v = (tid < 64) ? agg[(size_t)row * DFEAT + tid]
                             : xin[(size_t)row * DFEAT + (tid - 64)];
        As[r][tid] = v;
    }
    // --- stage B (concatenated weights, coalesced global reads) ---
    for (int i = tid; i < 64 * 64; i += 128) {
        int n = i >> 6, k = i & 63;
        Bs[n][k]      = Wrel[i];
        Bs[n][64 + k] = Wroot[i];
    }
    __syncthreads();

    const int wave = tid >> 5;
    const int lane = tid & 31;
    const int half = lane >> 4;
    const int l16  = lane & 15;
    const int ncol = wave * 16 + l16;

    v8f acc = {};
#pragma unroll
    for (int k0 = 0; k0 < 128; k0 += 4) {
        const int ka = k0 + half * 2;
        v2f a = *reinterpret_cast<const v2f*>(&As[l16][ka]);
        v2f b = *reinterpret_cast<const v2f*>(&Bs[ncol][ka]);
        acc = __builtin_amdgcn_wmma_f32_16x16x4_f32(
                  /*neg_a=*/false, a, /*neg_b=*/false, b,
                  /*c_mod=*/(short)0, acc, /*reuse_a=*/false, /*reuse_b=*/false);
    }

    const float bv = bias[ncol];
#pragma unroll
    for (int r = 0; r < 8; ++r) {
        int row = rb + r + half * 8;
        float v = acc[r] + bv;
        if (RELU) v = fmaxf(v, 0.0f);
        if (row < n_nodes) out[(size_t)row * DFEAT + ncol] = v;
    }
}

// ---------------------------------------------------------------------------
// Launcher: two GraphConv layers.
// inputs: x, edge_index, W1_rel, b1_rel, W1_root, W2_rel, b2_rel, W2_root
// ws: agg [N*64 f32] + h [N*64 f32]  (~51 MB)
// ---------------------------------------------------------------------------
extern "C" void kernel_launch(void* const* d_in, const int* in_sizes, int n_in,
                              void* d_out, int out_size, void* d_ws, size_t ws_size,
                              hipStream_t stream) {
    const float* x       = (const float*)d_in[0];
    const int*   edge    = (const int*)d_in[1];
    const float* W1_rel  = (const float*)d_in[2];
    const float* b1_rel  = (const float*)d_in[3];
    const float* W1_root = (const float*)d_in[4];
    const float* W2_rel  = (const float*)d_in[5];
    const float* b2_rel  = (const float*)d_in[6];
    const float* W2_root = (const float*)d_in[7];
    float* out = (float*)d_out;

    const int n_nodes = in_sizes[0] / DFEAT;   // 100000
    const int n_edges = in_sizes[1] / 2;       // 1200000
    const int* src = edge;
    const int* dst = edge + n_edges;

    float* agg = (float*)d_ws;
    float* h   = agg + (size_t)n_nodes * DFEAT;

    const int nElem = n_nodes * DFEAT;
    dim3 zblk(256), zgrd((nElem + 255) / 256);
    long long sthreads = (long long)n_edges * 16;
    dim3 sblk(256), sgrd((unsigned)((sthreads + 255) / 256));
    dim3 gblk(128), ggrd((n_nodes + 15) / 16);

    // Layer 1: h = relu([segment_sum(x) | x] @ [W1_rel;W1_root]^T + b1)
    gnn_zero_kernel<<<zgrd, zblk, 0, stream>>>(agg, nElem);
    gnn_scatter_add_kernel<<<sgrd, sblk, 0, stream>>>(x, src, dst, agg, n_edges);
    gnn_gconv_gemm_kernel<true><<<ggrd, gblk, 0, stream>>>(
        agg, x, W1_rel, W1_root, b1_rel, h, n_nodes);

    // Layer 2: out = [segment_sum(h) | h] @ [W2_rel;W2_root]^T + b2
    gnn_zero_kernel<<<zgrd, zblk, 0, stream>>>(agg, nElem);
    gnn_scatter_add_kernel<<<sgrd, sblk, 0, stream>>>(h, src, dst, agg, n_edges);
    gnn_gconv_gemm_kernel<false><<<ggrd, gblk, 0, stream>>>(
        agg, h, W2_rel, W2_root, b2_rel, out, n_nodes);
}